// BalanceLabels_86260123174045
// MI455X (gfx1250) — compile-verified
//
#include <hip/hip_runtime.h>
#include <stdint.h>

typedef __attribute__((ext_vector_type(4))) float v4f;
typedef int v4i __attribute__((vector_size(16)));  // builtin param pointee type

#define GAS __attribute__((address_space(1)))
#define LAS __attribute__((address_space(3)))

#define TPB 256
#define UNROLL 8
#define ACPOL 1  // TH_NT (single-touch 512 MB stream > 192 MB L2), WGP scope

// CDNA5 async LDS copy path: data moves HBM -> LDS -> HBM via the async
// data mover (ASYNCcnt), never passing through VGPRs.
#if defined(__HIP_DEVICE_COMPILE__) && defined(__gfx1250__) &&                 \
    __has_builtin(__builtin_amdgcn_global_load_async_to_lds_b128) &&           \
    __has_builtin(__builtin_amdgcn_global_store_async_from_lds_b128)
#define USE_ASYNC_LDS 1
#else
#define USE_ASYNC_LDS 0
#endif

#if defined(__HIP_DEVICE_COMPILE__) &&                                         \
    __has_builtin(__builtin_amdgcn_s_wait_asynccnt)
#define WAIT_ASYNC0() __builtin_amdgcn_s_wait_asynccnt(0)
#else
#define WAIT_ASYNC0() asm volatile("s_wait_asynccnt 0" ::: "memory")
#endif

#if USE_ASYNC_LDS
// Burst 8 independent async b128 loads for one tile (4 KB per wave in flight).
__device__ __forceinline__ void async_load_tile(const GAS v4f* g, LAS v4i* l) {
#pragma unroll
  for (int k = 0; k < UNROLL; ++k)
    __builtin_amdgcn_global_load_async_to_lds_b128(
        (GAS v4i*)(g + k * TPB), l + k * TPB, 0, ACPOL);
}
// Burst 8 async b128 stores straight from LDS back to HBM.
__device__ __forceinline__ void async_store_tile(GAS v4f* g, LAS v4i* l) {
#pragma unroll
  for (int k = 0; k < UNROLL; ++k)
    __builtin_amdgcn_global_store_async_from_lds_b128(
        (GAS v4i*)(g + k * TPB), l + k * TPB, 0, ACPOL);
}
#endif

__global__ __launch_bounds__(TPB) void BalanceLabels_copy_kernel(
    const float* __restrict__ src, float* __restrict__ dst,
    long long n, long long ntiles) {
  const int tid = threadIdx.x;
  const long long tileSize = (long long)TPB * UNROLL;  // float4 units per tile

#if USE_ASYNC_LDS
  // Two 32 KB stages -> double-buffered pipeline. Every WAIT_ASYNC0() is a
  // full drain, so buffer reuse is race-free by construction; in steady state
  // each drain-to-drain window has 8 stores + 8 loads (8 KB/wave) in flight.
  __shared__ v4f lbuf[2 * TPB * UNROLL];  // 64 KB -> 5 blocks / 320 KB WGP
  LAS v4i* stage0 = (LAS v4i*)lbuf + tid;
  LAS v4i* stage1 = stage0 + TPB * UNROLL;
  const GAS v4f* gs = (const GAS v4f*)src;
  GAS v4f* gd = (GAS v4f*)dst;

  const long long stride = gridDim.x;
  long long tload = blockIdx.x;
  long long tstore = tload;
  int cur = 0;
  if (tload < ntiles) async_load_tile(gs + tload * tileSize + tid, stage0);
  tload += stride;

  while (tstore < ntiles) {
    WAIT_ASYNC0();  // drain: current stage's loads (and any prior stores) done
    async_store_tile(gd + tstore * tileSize + tid, cur ? stage1 : stage0);
    if (tload < ntiles)
      async_load_tile(gs + tload * tileSize + tid, cur ? stage0 : stage1);
    cur ^= 1;
    tstore = tload;
    tload += stride;
  }
  WAIT_ASYNC0();  // drain final store burst
#else
  // Fallback: unrolled non-temporal b128 register copy.
  const v4f* gs = (const v4f*)src;
  v4f* gd = (v4f*)dst;
  for (long long t = blockIdx.x; t < ntiles; t += gridDim.x) {
    const long long base = t * tileSize + tid;
    v4f r[UNROLL];
#pragma unroll
    for (int k = 0; k < UNROLL; ++k)
      r[k] = __builtin_nontemporal_load(gs + base + (long long)k * TPB);
#pragma unroll
    for (int k = 0; k < UNROLL; ++k)
      __builtin_nontemporal_store(r[k], gd + base + (long long)k * TPB);
  }
#endif

  // Scalar remainder (elements beyond the last full tile). For the reference
  // shape (2^26 floats) this loop body never executes.
  const long long covered = ntiles * tileSize * 4;
  if (blockIdx.x == 0) {
    for (long long j = covered + tid; j < n; j += TPB) dst[j] = src[j];
  }
}

extern "C" void kernel_launch(void* const* d_in, const int* in_sizes, int n_in,
                              void* d_out, int out_size, void* d_ws,
                              size_t ws_size, hipStream_t stream) {
  (void)in_sizes; (void)n_in; (void)d_ws; (void)ws_size;
  // Reference output == inputs (forward of ScaleGrad-style module); `target`
  // only feeds the (discarded) gradient weights, so we never read it.
  const float* src = (const float*)d_in[0];
  float* dst = (float*)d_out;

  const long long n = (long long)out_size;           // fp32 elements
  const long long n4 = n / 4;                        // float4 elements
  const long long tileSize = (long long)TPB * UNROLL;
  const long long ntiles = n4 / tileSize;            // 8192 for 2^26 floats

  // ~4 tiles per block so the double-buffer pipeline has a steady state;
  // 2048 blocks x 8 wave32 = 16K waves, ample HBM-level parallelism.
  long long nblocks = (ntiles + 3) / 4;
  if (nblocks < 1) nblocks = 1;
  if (nblocks > 2048) nblocks = 2048;
  dim3 grid((unsigned)nblocks), block(TPB);
  BalanceLabels_copy_kernel<<<grid, block, 0, stream>>>(src, dst, n, ntiles);
}